// KMeansPlusPlus_11493332484394
// MI455X (gfx1250) — compile-verified
//
#include <hip/hip_runtime.h>

typedef __attribute__((ext_vector_type(2))) float v2f;
typedef __attribute__((ext_vector_type(8))) float v8f;

#define NPTS      200000
#define DIMS      64
#define KC        256
#define NPAIRS    (NPTS / 32)       // 6250 32-point pair-tiles (divides exactly)
#define CENT_PAD  68                // padded LDS row stride (floats) -> conflict-free b64 reads
#define ABLOCKS   128
#define WPB       8                 // waves per block (256 threads, wave32)

// ---------------------------------------------------------------------------
// init: copy initial centroids into workspace, clear done flag
// ---------------------------------------------------------------------------
__global__ void kinit(const float* __restrict__ cIn, float* __restrict__ cent,
                      int* __restrict__ done) {
    int i = blockIdx.x * blockDim.x + threadIdx.x;
    if (i < KC * DIMS) cent[i] = cIn[i];
    if (i == 0) *done = 0;
}

// ---------------------------------------------------------------------------
// prep (per iteration): csq[k] = |c_k|^2 ; zero sums/counts
// ---------------------------------------------------------------------------
__global__ __launch_bounds__(256) void kprep(const float* __restrict__ cent,
                                             float* __restrict__ csq,
                                             float* __restrict__ sums,
                                             float* __restrict__ counts) {
    int t = threadIdx.x;
    float s = 0.0f;
    #pragma unroll 8
    for (int d = 0; d < DIMS; ++d) {
        float v = cent[t * DIMS + d];
        s += v * v;
    }
    csq[t] = s;
    counts[t] = 0.0f;
    for (int i = t; i < KC * DIMS; i += 256) sums[i] = 0.0f;
}

// ---------------------------------------------------------------------------
// assign: dual-chain WMMA fp32 distance tiles -> argmin -> LDS segment-sum
// ---------------------------------------------------------------------------
__global__ __launch_bounds__(256) void kassign(const float* __restrict__ data,
                                               const float* __restrict__ gcent,
                                               const float* __restrict__ gcsq,
                                               float* __restrict__ gsums,
                                               float* __restrict__ gcounts,
                                               int* __restrict__ out,
                                               const int* __restrict__ done) {
    if (*done) return;  // uniform early-out once converged

    __shared__ float s_cent[KC * CENT_PAD];   // padded centroid tile
    __shared__ float s_sums[KC * DIMS];       // per-block segment sums
    __shared__ float s_counts[KC];
    __shared__ float s_csq[KC];
    __shared__ int   s_stage[WPB][32];        // per-wave assignment staging

    const int tid = threadIdx.x;
    for (int i = tid; i < KC * DIMS; i += 256) {
        int k = i >> 6, d = i & 63;
        s_cent[k * CENT_PAD + d] = gcent[i];
        s_sums[i] = 0.0f;
    }
    if (tid < KC) { s_counts[tid] = 0.0f; s_csq[tid] = gcsq[tid]; }
    __syncthreads();

    const int lane  = tid & 31;
    const int wave  = tid >> 5;
    const int n15   = lane & 15;
    const int half  = lane >> 4;      // 0: K={0,1}, 1: K={2,3} within each 4-chunk
    const int koff  = half * 2;
    const int gwave = blockIdx.x * WPB + wave;
    const int nwav  = gridDim.x * WPB;

    for (int t = gwave; t < NPAIRS; t += nwav) {
        const int base = t * 32;      // 32 consecutive points: tiles [base,+16) [base+16,+32)

        // ---- A fragments for both point tiles (reused across all 16 colTiles) ----
        const float* arow0 = data + (size_t)(base + n15) * DIMS + koff;
        const float* arow1 = arow0 + 16 * DIMS;
        v2f a0[16], a1[16];
        #pragma unroll
        for (int j = 0; j < 16; ++j) {
            a0[j] = *(const v2f*)(arow0 + 4 * j);
            a1[j] = *(const v2f*)(arow1 + 4 * j);
        }

        float bd0[8], bd1[8];
        int   bi0[8], bi1[8];
        #pragma unroll
        for (int r = 0; r < 8; ++r) {
            bd0[r] = 3.4e38f; bi0[r] = 0;
            bd1[r] = 3.4e38f; bi1[r] = 0;
        }

        #pragma unroll 1
        for (int ct = 0; ct < 16; ++ct) {
            const int c0 = ct * 16;
            const float* brow = s_cent + (size_t)(c0 + n15) * CENT_PAD + koff;
            v8f acc0 = {};
            v8f acc1 = {};
            // two independent WMMA chains share each B fragment
            #pragma unroll
            for (int j = 0; j < 16; ++j) {
                v2f b = *(const v2f*)(brow + 4 * j);
                acc0 = __builtin_amdgcn_wmma_f32_16x16x4_f32(
                           false, a0[j], false, b, (short)0, acc0, false, false);
                acc1 = __builtin_amdgcn_wmma_f32_16x16x4_f32(
                           false, a1[j], false, b, (short)0, acc1, false, false);
            }
            const float cq = s_csq[c0 + n15];
            const int   ci = c0 + n15;
            #pragma unroll
            for (int r = 0; r < 8; ++r) {
                float d0 = cq - 2.0f * acc0[r];   // argmin-equivalent distance
                float d1 = cq - 2.0f * acc1[r];
                if (d0 < bd0[r]) { bd0[r] = d0; bi0[r] = ci; }
                if (d1 < bd1[r]) { bd1[r] = d1; bi1[r] = ci; }
            }
        }

        // ---- argmin across the 16 lanes of each half (rows 0-7 / 8-15) ----
        #pragma unroll
        for (int s = 8; s >= 1; s >>= 1) {
            #pragma unroll
            for (int r = 0; r < 8; ++r) {
                float od0 = __shfl_xor(bd0[r], s, 32);
                int   oi0 = __shfl_xor(bi0[r], s, 32);
                if (od0 < bd0[r] || (od0 == bd0[r] && oi0 < bi0[r])) {
                    bd0[r] = od0; bi0[r] = oi0;
                }
                float od1 = __shfl_xor(bd1[r], s, 32);
                int   oi1 = __shfl_xor(bi1[r], s, 32);
                if (od1 < bd1[r] || (od1 == bd1[r] && oi1 < bi1[r])) {
                    bd1[r] = od1; bi1[r] = oi1;
                }
            }
        }

        // stage: lane0 -> rows 0-7 of each tile, lane16 -> rows 8-15
        if (lane == 0 || lane == 16) {
            #pragma unroll
            for (int r = 0; r < 8; ++r) {
                s_stage[wave][half * 8 + r]      = bi0[r];
                s_stage[wave][16 + half * 8 + r] = bi1[r];
            }
        }
        asm volatile("s_wait_dscnt 0x0" ::: "memory");

        // all 32 lanes write one assignment each
        out[base + lane] = s_stage[wave][lane];

        // ---- segment-sum accumulation into LDS (banks 2l,2l+1: conflict-free) ----
        #pragma unroll 1
        for (int p = 0; p < 32; ++p) {
            const int aidx = s_stage[wave][p];
            v2f x = *(const v2f*)(data + (size_t)(base + p) * DIMS + 2 * lane);
            unsafeAtomicAdd(&s_sums[aidx * DIMS + 2 * lane],     x.x);
            unsafeAtomicAdd(&s_sums[aidx * DIMS + 2 * lane + 1], x.y);
            if (lane == 0) unsafeAtomicAdd(&s_counts[aidx], 1.0f);
        }
    }
    __syncthreads();

    // ---- flush per-block partials to global (skip zeros to cut atomics) ----
    for (int i = tid; i < KC * DIMS; i += 256)
        if (s_sums[i] != 0.0f) unsafeAtomicAdd(&gsums[i], s_sums[i]);
    if (tid < KC && s_counts[tid] != 0.0f) unsafeAtomicAdd(&gcounts[tid], s_counts[tid]);
}

// ---------------------------------------------------------------------------
// update: new centroids, max movement, convergence / max_iter flag
// ---------------------------------------------------------------------------
__global__ __launch_bounds__(256) void kupdate(float* __restrict__ cent,
                                               const float* __restrict__ sums,
                                               const float* __restrict__ counts,
                                               int* __restrict__ done,
                                               const int* __restrict__ maxit,
                                               int iter) {
    __shared__ float red[256];
    if (*done) return;  // uniform
    int k = threadIdx.x;
    float cnt = counts[k];
    float m2 = 0.0f;
    if (cnt > 0.0f) {
        float inv = 1.0f / cnt;
        #pragma unroll 8
        for (int d = 0; d < DIMS; ++d) {
            float oldv = cent[k * DIMS + d];
            float newv = sums[k * DIMS + d] * inv;
            float df = newv - oldv;
            m2 += df * df;
            cent[k * DIMS + d] = newv;
        }
    }
    red[k] = m2;
    __syncthreads();
    for (int s = 128; s >= 1; s >>= 1) {
        if (k < s) red[k] = fmaxf(red[k], red[k + s]);
        __syncthreads();
    }
    if (k == 0) {
        int stop = (red[0] < 0.01f) || ((iter + 1) >= *maxit);
        *done = stop ? 1 : 0;
    }
}

// ---------------------------------------------------------------------------
extern "C" void kernel_launch(void* const* d_in, const int* in_sizes, int n_in,
                              void* d_out, int out_size, void* d_ws, size_t ws_size,
                              hipStream_t stream) {
    const float* data  = (const float*)d_in[0];   // [N, D] fp32
    const float* c0    = (const float*)d_in[1];   // [K, D] fp32
    const int*   maxit = (const int*)d_in[2];     // scalar (=10)

    float* ws     = (float*)d_ws;                 // needs ~133 KB
    float* cent   = ws;                           // K*D
    float* sums   = ws + KC * DIMS;               // K*D
    float* csq    = ws + 2 * KC * DIMS;           // K
    float* counts = csq + KC;                     // K
    int*   done   = (int*)(counts + KC);          // 1
    int*   out    = (int*)d_out;                  // [N] int32 assignments

    kinit<<<(KC * DIMS + 255) / 256, 256, 0, stream>>>(c0, cent, done);
    for (int it = 0; it < 10; ++it) {
        kprep<<<1, 256, 0, stream>>>(cent, csq, sums, counts);
        kassign<<<ABLOCKS, 256, 0, stream>>>(data, cent, csq, sums, counts, out, done);
        kupdate<<<1, 256, 0, stream>>>(cent, sums, counts, done, maxit, it);
    }
}